// triplet_loss_6141803233797
// MI455X (gfx1250) — compile-verified
//
#include <hip/hip_runtime.h>
#include <math.h>

// Problem constants (from reference setup_inputs)
#define B_   64
#define C_   768
#define S_   576          // 24*24
#define F_   2
#define N_   128          // 2B rows of feats
#define D_   2304         // blocks*C
#define GSZ  192          // S/blocks
#define NCH  12           // channel chunks for heat partials
#define CCH  (C_/NCH)     // 64 channels per chunk

typedef __attribute__((ext_vector_type(2))) float v2f;
typedef __attribute__((ext_vector_type(8))) float v8f;

// Workspace layout (float units). heat_part region is reused for X/sq/dist
// after it has been consumed (kernels are stream-ordered).
#define OFF_X    0                        // N*D = 294912 floats (reuses part region)
#define OFF_SQ   (N_*D_)                  // 128 floats
#define OFF_DIST (OFF_SQ + 128)           // N*N = 16384 floats
#define OFF_HEAT ((size_t)F_*B_*NCH*S_)   // 884736 : heat F*B*S
#define OFF_GRP  (OFF_HEAT + (size_t)F_*B_*S_) // int region F*B*S

// ---------------------------------------------------------------- pass 1a:
// partial channel sums for heat: part[f][b][ch][s] = sum over 64 channels
__global__ void k_heat_part(const float* __restrict__ x1,
                            const float* __restrict__ x2,
                            float* __restrict__ part) {
  int s  = threadIdx.x;          // 0..575, coalesced
  int ch = blockIdx.x;           // 0..11
  int b  = blockIdx.y;
  int f  = blockIdx.z;
  const float* xf = f ? x2 : x1;
  const float* p = xf + ((size_t)b * C_ + (size_t)ch * CCH) * S_ + s;
  float acc = 0.f;
  #pragma unroll 8
  for (int cc = 0; cc < CCH; ++cc) acc += p[(size_t)cc * S_];
  part[(((size_t)f * B_ + b) * NCH + ch) * S_ + s] = acc;
}

// heat[f][b][s] = mean over C (deterministic fixed-order reduce)
__global__ void k_heat_reduce(const float* __restrict__ part,
                              float* __restrict__ heat) {
  int s = threadIdx.x, b = blockIdx.x, f = blockIdx.y;
  const float* p = part + ((size_t)f * B_ + b) * NCH * S_ + s;
  float acc = 0.f;
  #pragma unroll
  for (int i = 0; i < NCH; ++i) acc += p[(size_t)i * S_];
  heat[((size_t)f * B_ + b) * S_ + s] = acc * (1.0f / C_);
}

// ---------------------------------------------------------------- ranking:
// stable descending rank (matches jnp.argsort(-heat)); group = rank / 192
__global__ void k_rank(const float* __restrict__ heat, int* __restrict__ grp) {
  __shared__ float h[S_];
  int s = threadIdx.x, b = blockIdx.x, f = blockIdx.y;
  const float* hp = heat + ((size_t)f * B_ + b) * S_;
  h[s] = hp[s];
  __syncthreads();
  float mine = h[s];
  int r = 0;
  for (int j = 0; j < S_; ++j) {
    float v = h[j];
    r += (int)((v > mine) || ((v == mine) && (j < s)));
  }
  grp[((size_t)f * B_ + b) * S_ + s] = r / GSZ;   // 0,1,2
}

// ---------------------------------------------------------------- pass 2:
// per-(f,b,c) wave accumulates the 3 rank-group means; coalesced s-stream
__global__ void k_pool(const float* __restrict__ x1,
                       const float* __restrict__ x2,
                       const int* __restrict__ grp,
                       float* __restrict__ X) {
  int tid  = blockIdx.x * blockDim.x + threadIdx.x;
  int wid  = tid >> 5;
  int lane = tid & 31;
  int f    = wid / (B_ * C_);
  int rem  = wid - f * (B_ * C_);
  int b    = rem / C_;
  int c    = rem - b * C_;
  const float* xf = f ? x2 : x1;
  const float* xp = xf + ((size_t)b * C_ + c) * S_;
  const int*   gp = grp + ((size_t)f * B_ + b) * S_;
  float a0 = 0.f, a1 = 0.f, a2 = 0.f;
  #pragma unroll
  for (int i = 0; i < S_ / 32; ++i) {
    int s = lane + (i << 5);
    float v = xp[s];
    int g = gp[s];
    a0 += (g == 0) ? v : 0.f;
    a1 += (g == 1) ? v : 0.f;
    a2 += (g == 2) ? v : 0.f;
  }
  #pragma unroll
  for (int off = 16; off >= 1; off >>= 1) {
    a0 += __shfl_xor(a0, off, 32);
    a1 += __shfl_xor(a1, off, 32);
    a2 += __shfl_xor(a2, off, 32);
  }
  if (lane == 0) {
    int row = f * B_ + b;
    const float inv = 1.0f / GSZ;
    X[(size_t)row * D_ + 0 * C_ + c] = a0 * inv;   // block-major layout
    X[(size_t)row * D_ + 1 * C_ + c] = a1 * inv;
    X[(size_t)row * D_ + 2 * C_ + c] = a2 * inv;
  }
}

// ---------------------------------------------------------------- norms:
// b /= clip(||b||,1e-12)  then  x = b/(||b||+1e-6); also sq[row] = ||x||^2
__global__ void k_norm(float* __restrict__ X, float* __restrict__ sq) {
  __shared__ float red[256];
  int row = blockIdx.x, t = threadIdx.x;
  float* xp = X + (size_t)row * D_;
  float v[D_ / 256];
  float ss = 0.f;
  #pragma unroll
  for (int k = 0; k < D_ / 256; ++k) { v[k] = xp[t + 256 * k]; ss += v[k] * v[k]; }
  red[t] = ss; __syncthreads();
  for (int off = 128; off >= 1; off >>= 1) {
    if (t < off) red[t] += red[t + off];
    __syncthreads();
  }
  float inv1 = 1.0f / fmaxf(sqrtf(red[0]), 1e-12f);
  __syncthreads();
  float ss2 = 0.f;
  #pragma unroll
  for (int k = 0; k < D_ / 256; ++k) { v[k] *= inv1; ss2 += v[k] * v[k]; }
  red[t] = ss2; __syncthreads();
  for (int off = 128; off >= 1; off >>= 1) {
    if (t < off) red[t] += red[t + off];
    __syncthreads();
  }
  float n2sq = red[0];
  float inv2 = 1.0f / (sqrtf(n2sq) + 1e-6f);
  #pragma unroll
  for (int k = 0; k < D_ / 256; ++k) xp[t + 256 * k] = v[k] * inv2;
  if (t == 0) sq[row] = n2sq * inv2 * inv2;
}

// ---------------------------------------------------------------- WMMA Gram:
// one wave per 16x16 tile of G = X*X^T (f32, K=4 per v_wmma_f32_16x16x4_f32),
// epilogue converts to clipped Euclidean distance.
__global__ void k_dist_wmma(const float* __restrict__ X,
                            const float* __restrict__ sq,
                            float* __restrict__ dist) {
  int tid  = blockIdx.x * blockDim.x + threadIdx.x;
  int wave = tid >> 5;            // 0..63 => 8x8 tile grid
  int lane = tid & 31;
  int ti = wave >> 3, tj = wave & 7;
  int hh = lane >> 4;             // half-wave: K split (A) / K split (B)
  int mn = lane & 15;             // M (A) / N (B)
  const float* pa = X + (size_t)(ti * 16 + mn) * D_ + hh * 2;
  const float* pb = X + (size_t)(tj * 16 + mn) * D_ + hh * 2;
  v8f acc = {0.f, 0.f, 0.f, 0.f, 0.f, 0.f, 0.f, 0.f};
  for (int kb = 0; kb < D_; kb += 4) {
    v2f a = *(const v2f*)(pa + kb);   // A[M=mn][kb + 2*hh + {0,1}]
    v2f b = *(const v2f*)(pb + kb);   // B[kb + 2*hh + {0,1}][N=mn] (B = X^T tile)
    acc = __builtin_amdgcn_wmma_f32_16x16x4_f32(
        /*neg_a=*/false, a, /*neg_b=*/false, b,
        /*c_mod=*/(short)0, acc, /*reuse_a=*/false, /*reuse_b=*/false);
  }
  #pragma unroll
  for (int v = 0; v < 8; ++v) {
    int gi = ti * 16 + v + hh * 8;    // C/D layout: M = vgpr + 8*(lane>=16)
    int gj = tj * 16 + mn;
    float d2 = sq[gi] + sq[gj] - 2.0f * acc[v];
    dist[gi * N_ + gj] = sqrtf(fmaxf(d2, 1e-6f));
  }
}

// ---------------------------------------------------------------- mining+loss
__global__ void k_loss(const float* __restrict__ dist,
                       const int* __restrict__ labels,
                       float* __restrict__ out) {
  __shared__ float red[N_];
  int i = threadIdx.x;              // 0..127
  int li = labels[i & 63];
  int base = (i < 64) ? 64 : 0;     // opposite half only
  float ap = -1e30f, an = 1e30f;
  const float* dr = dist + i * N_;
  for (int j = 0; j < 64; ++j) {
    int jj = base + j;
    float d = dr[jj];
    bool same = (labels[jj & 63] == li);
    ap = same ? fmaxf(ap, d) : ap;
    an = same ? an : fminf(an, d);
  }
  red[i] = fmaxf(ap - an + 0.3f, 0.0f);
  __syncthreads();
  for (int off = 64; off >= 1; off >>= 1) {
    if (i < off) red[i] += red[i + off];
    __syncthreads();
  }
  if (i == 0) out[0] = red[0] * (1.0f / N_);
}

extern "C" void kernel_launch(void* const* d_in, const int* in_sizes, int n_in,
                              void* d_out, int out_size, void* d_ws, size_t ws_size,
                              hipStream_t stream) {
  const float* x1     = (const float*)d_in[0];
  const float* x2     = (const float*)d_in[1];
  const int*   labels = (const int*)d_in[2];
  // d_in[3] = blocks (==3 per setup_inputs; compiled in)

  float* ws   = (float*)d_ws;
  float* part = ws;                 // reused as X/sq/dist once consumed
  float* heat = ws + OFF_HEAT;
  int*   grp  = (int*)(ws + OFF_GRP);
  float* X    = ws + OFF_X;
  float* sq   = ws + OFF_SQ;
  float* dist = ws + OFF_DIST;

  k_heat_part  <<<dim3(NCH, B_, F_), S_, 0, stream>>>(x1, x2, part);
  k_heat_reduce<<<dim3(B_, F_),       S_, 0, stream>>>(part, heat);
  k_rank       <<<dim3(B_, F_),       S_, 0, stream>>>(heat, grp);
  k_pool       <<<(F_ * B_ * C_ * 32) / 256, 256, 0, stream>>>(x1, x2, grp, X);
  k_norm       <<<N_, 256, 0, stream>>>(X, sq);
  k_dist_wmma  <<<8, 256, 0, stream>>>(X, sq, dist);
  k_loss       <<<1, N_, 0, stream>>>(dist, labels, (float*)d_out);
}